// QuantizationLayer_57183194579680
// MI455X (gfx1250) — compile-verified
//
#include <hip/hip_runtime.h>

#define NUM_BINS 5
#define HEIGHT 260
#define WIDTH 346

typedef float v2f __attribute__((ext_vector_type(2)));
typedef float v8f __attribute__((ext_vector_type(8)));

// D = A(16x4 f32) * B(4x16 f32) + C(16x16 f32)
#define WMMA_F32(a, b, c) \
    __builtin_amdgcn_wmma_f32_16x16x4_f32(false, (a), false, (b), (short)0, (c), false, false)

// v_permlane16_swap_b32: a.hi16 <-> b.lo16.
// In:  a = [aL | aH],  b = [bL | bH]
// Out: a = [aL | bL],  b = [aH | bH]
__device__ __forceinline__ void permswap16(float& a, float& b) {
    asm("v_permlane16_swap_b32 %0, %1" : "+v"(a), "+v"(b));
}

// Exact leaky-relu in 2 VALU ops: max_num(0.1x, x) (no canonicalize).
__device__ __forceinline__ float leaky(float x) {
    float y = 0.1f * x, r;
    asm("v_max_num_f32 %0, %1, %2" : "=v"(r) : "v"(x), "v"(y));
    return r;
}

// Convert activation tiles d[3] (C/D layout: M=channel, N=event) into the 10
// K-step B tiles (K=channel, N=event), destroying d in place. Each
// permlane16_swap produces TWO B vgprs:
//   swap(d[t][4p+q], d[t][4p+q+2]) -> B[4t+p][q] and B[4t+p+2][q]
// Tile 2's byproducts (channels 40-47) are dead padding.
__device__ __forceinline__ void buildB(v8f* d, v2f* B) {
#pragma unroll
    for (int t = 0; t < 2; ++t)
#pragma unroll
        for (int p = 0; p < 2; ++p)
#pragma unroll
            for (int q = 0; q < 2; ++q) {
                float a = d[t][4 * p + q];
                float b = d[t][4 * p + q + 2];
                permswap16(a, b);
                B[4 * t + p][q] = a;
                B[4 * t + p + 2][q] = b;
            }
#pragma unroll
    for (int p = 0; p < 2; ++p)
#pragma unroll
        for (int q = 0; q < 2; ++q) {
            float a = d[2][4 * p + q];
            float b = d[2][4 * p + q + 2];
            permswap16(a, b);
            B[8 + p][q] = a;   // byproduct b holds channels 40+ (unused)
        }
}

__global__ void __launch_bounds__(256) voxel_zero_kernel(float* p, int n) {
    int i = blockIdx.x * blockDim.x + threadIdx.x;
    if (i < n) p[i] = 0.0f;
}

__global__ void __launch_bounds__(256) voxel_mlp_kernel(
    const float* __restrict__ events,
    const float* __restrict__ w1, const float* __restrict__ b1,
    const float* __restrict__ w2, const float* __restrict__ b2,
    const float* __restrict__ w3, const float* __restrict__ b3,
    float* __restrict__ voxel, int n_events)
{
    const int lane = threadIdx.x & 31;
    const int m = lane & 15;
    const bool hi = lane >= 16;

    float t0 = events[0];
    float tN = events[(size_t)(n_events - 1) * 4];
    float dT = tN - t0;
    if (dT == 0.0f) dT = 1.0f;
    float invdT = 1.0f / dT;

    // ---- Preload loop-invariant A-tiles (weights) and bias C-tiles ----
    v2f a1t[3];
#pragma unroll
    for (int t = 0; t < 3; ++t) {
        int o = t * 16 + m;
        float x0 = 0.0f, x1 = 0.0f;
        if (o < 40) {
            int c0 = hi ? 2 : 0;
            x0 = w1[o * 4 + c0];
            x1 = w1[o * 4 + c0 + 1];
        }
        a1t[t][0] = x0; a1t[t][1] = x1;
    }
    v8f bias1[3], bias2[3];
#pragma unroll
    for (int t = 0; t < 3; ++t) {
#pragma unroll
        for (int r = 0; r < 8; ++r) {
            int ch = t * 16 + r + (hi ? 8 : 0);
            bias1[t][r] = (ch < 40) ? b1[ch] : 0.0f;
            bias2[t][r] = (ch < 40) ? b2[ch] : 0.0f;
        }
    }
    v2f a2t[3][10];
#pragma unroll
    for (int t = 0; t < 3; ++t) {
        int o = t * 16 + m;
#pragma unroll
        for (int j = 0; j < 10; ++j) {
            float x0 = 0.0f, x1 = 0.0f;
            if (o < 40) {
                int k = 4 * j + (hi ? 2 : 0);
                x0 = w2[o * 40 + k];
                x1 = w2[o * 40 + k + 1];
            }
            a2t[t][j][0] = x0; a2t[t][j][1] = x1;
        }
    }
    v2f a3t[10];
#pragma unroll
    for (int j = 0; j < 10; ++j) {
        float x0 = 0.0f, x1 = 0.0f;
        if (m == 0) {
            int k = 4 * j + (hi ? 2 : 0);
            x0 = w3[k];
            x1 = w3[k + 1];
        }
        a3t[j][0] = x0; a3t[j][1] = x1;
    }
    // Layer-3 C init: b3 on row M=0 (vgpr 0, lanes 0-15), zero elsewhere.
    v8f c3init = { 0, 0, 0, 0, 0, 0, 0, 0 };
    c3init[0] = hi ? 0.0f : b3[0];

    int wave = (int)((blockIdx.x * blockDim.x + threadIdx.x) >> 5);
    int nwaves = (int)((gridDim.x * blockDim.x) >> 5);
    int chunks_per_bin = (n_events + 15) >> 4;

    auto loadEv = [&](int cb, bool& valid, int& xi, int& yi, float4& ev) {
        int e = cb * 16 + m;
        valid = e < n_events;
        int ecl = valid ? e : (n_events - 1);
        ev = reinterpret_cast<const float4*>(events)[ecl];
        xi = (int)ev.y;
        yi = (int)ev.z;
    };

    for (int bin = 0; bin < NUM_BINS; ++bin) {
        const float bc = (float)bin * (1.0f / (NUM_BINS - 1));
        const int voxBase = WIDTH * HEIGHT * bin;

        if (wave >= chunks_per_bin) continue;
        int xi, yi;
        bool valid;
        float4 ev;
        loadEv(wave, valid, xi, yi, ev);

        for (int cb = wave; cb < chunks_per_bin; cb += nwaves) {
            // Software pipeline: issue next chunk's event load now.
            int cn = cb + nwaves;
            int xiN, yiN;
            bool validN;
            float4 evN;
            loadEv(cn < chunks_per_bin ? cn : cb, validN, xiN, yiN, evN);

            float ts = (ev.x - t0) * invdT;
            float pol = (ev.w == 0.0f) ? -1.0f : ev.w;

            // Input B tile (4 features x 16 events).
            v2f B1;
            B1[0] = hi ? (ev.y * (1.0f / WIDTH)) : (ts - bc);
            B1[1] = hi ? (ev.z * (1.0f / HEIGHT)) : pol;

            // Layer 1: one K-step, C = bias tiles.
            v8f d1[3];
#pragma unroll
            for (int t = 0; t < 3; ++t)
                d1[t] = WMMA_F32(a1t[t], B1, bias1[t]);
#pragma unroll
            for (int t = 0; t < 3; ++t)
#pragma unroll
                for (int r = 0; r < 8; ++r)
                    d1[t][r] = leaky(d1[t][r]);

            // Layer 2: relayout via 12 permlane16_swaps, then 10 K-steps.
            v2f B2[10];
            buildB(d1, B2);
            v8f d2[3];
#pragma unroll
            for (int t = 0; t < 3; ++t)
                d2[t] = WMMA_F32(a2t[t][0], B2[0], bias2[t]);
#pragma unroll
            for (int j = 1; j < 10; ++j)
#pragma unroll
                for (int t = 0; t < 3; ++t)
                    d2[t] = WMMA_F32(a2t[t][j], B2[j], d2[t]);
#pragma unroll
            for (int t = 0; t < 3; ++t)
#pragma unroll
                for (int r = 0; r < 8; ++r)
                    d2[t][r] = leaky(d2[t][r]);

            // Layer 3: same relayout, 10 K-steps into one tile (row 0 = v).
            v2f B3[10];
            buildB(d2, B3);
            v8f d3 = WMMA_F32(a3t[0], B3[0], c3init);
#pragma unroll
            for (int j = 1; j < 10; ++j)
                d3 = WMMA_F32(a3t[j], B3[j], d3);

            // v for event n sits at lane n (0-15), vgpr 0 (M=0, N=lane).
            if (!hi && valid) {
                float val = ts * d3[0];
                atomicAdd(&voxel[voxBase + WIDTH * yi + xi], val);
            }

            valid = validN; xi = xiN; yi = yiN; ev = evN;
        }
    }
}

extern "C" void kernel_launch(void* const* d_in, const int* in_sizes, int n_in,
                              void* d_out, int out_size, void* d_ws, size_t ws_size,
                              hipStream_t stream) {
    const float* events = (const float*)d_in[0];
    const float* w1 = (const float*)d_in[1];
    const float* b1 = (const float*)d_in[2];
    const float* w2 = (const float*)d_in[3];
    const float* b2 = (const float*)d_in[4];
    const float* w3 = (const float*)d_in[5];
    const float* b3 = (const float*)d_in[6];
    float* out = (float*)d_out;

    int n_events = in_sizes[0] / 4;

    voxel_zero_kernel<<<(out_size + 255) / 256, 256, 0, stream>>>(out, out_size);
    voxel_mlp_kernel<<<1024, 256, 0, stream>>>(events, w1, b1, w2, b2, w3, b3,
                                               out, n_events);
}